// MemoryModule_52398601011836
// MI455X (gfx1250) — compile-verified
//
#include <hip/hip_runtime.h>

typedef __attribute__((ext_vector_type(2))) float v2f;
typedef __attribute__((ext_vector_type(8))) float v8f;
typedef __attribute__((ext_vector_type(4))) unsigned int u32x4;
typedef __attribute__((ext_vector_type(8))) int i32x8;
typedef __attribute__((ext_vector_type(4))) int i32x4;

#define B_    32
#define C_    512
#define T_    1024
#define M_    512
#define N_    (B_ * T_)
#define ROWS  16
#define SSP   513            // padded LDS stride for score matrix
#define EPSF  1e-12f

#define HAVE_TDM (__has_builtin(__builtin_amdgcn_tensor_load_to_lds) && \
                  __has_builtin(__builtin_amdgcn_tensor_store_from_lds))

// Build a 2D TDM descriptor: tile of ROWS contiguous f32 elements per line,
// C_ lines, line stride T_ elements, into/from contiguous LDS.
__device__ __forceinline__ u32x4 tdm_group0(unsigned lds_off, const void* gptr) {
    unsigned long long ga = (unsigned long long)(uintptr_t)gptr;
    u32x4 g0;
    g0[0] = 1u;                                           // count=1, user 2D D#
    g0[1] = lds_off;                                      // LDS byte address
    g0[2] = (unsigned)ga;                                 // global addr [31:0]
    g0[3] = (unsigned)((ga >> 32) & 0x01FFFFFFu) | 0x80000000u; // [56:32] | type=2
    return g0;
}
__device__ __forceinline__ i32x8 tdm_group1() {
    i32x8 g1 = {};
    g1[0] = (2 << 16);                   // data_size=2 (4 bytes); no mask/pad
    g1[1] = (int)(((unsigned)T_ & 0xFFFFu) << 16);  // tensor_dim0 lo16 @bits[63:48]
    g1[2] = (int)(((unsigned)C_ & 0xFFFFu) << 16);  // tensor_dim1 lo16 @bits[95:80]
    g1[3] = (int)((unsigned)ROWS << 16); // tile_dim0=16 @bits[127:112]
    g1[4] = (int)C_;                     // tile_dim1=512 @bits[143:128]
    g1[5] = (int)T_;                     // tensor_dim0_stride lo32 @bits[191:160]
    g1[6] = 0; g1[7] = 0;
    return g1;
}

// ---------------------------------------------------------------------------
// Prep: per-memory-row squared norm + reciprocal norm; zero loss accumulators.
// ---------------------------------------------------------------------------
__global__ void kPrep(const float* __restrict__ mem,
                      float* __restrict__ m2, float* __restrict__ rm,
                      float* __restrict__ fea_acc, float* __restrict__ sep_acc) {
    int m = blockIdx.x * 256 + threadIdx.x;
    if (m < M_) {
        const float* row = mem + (size_t)m * C_;
        float s = 0.f;
        for (int c = 0; c < C_; ++c) s += row[c] * row[c];
        m2[m] = s;
        rm[m] = 1.0f / fmaxf(sqrtf(s), EPSF);
    }
    if (m == 0) { *fea_acc = 0.f; *sep_acc = 0.f; }
}

// ---------------------------------------------------------------------------
// Separation loss: sum of squared upper-triangular gram entries.
// ---------------------------------------------------------------------------
__global__ void kSep(const float* __restrict__ mem, float* __restrict__ sep_acc) {
    int idx = blockIdx.x * 256 + threadIdx.x;       // 0 .. 512*512-1
    int i = idx >> 9, j = idx & (M_ - 1);
    if (j <= i) return;
    const float* a = mem + (size_t)i * C_;
    const float* b = mem + (size_t)j * C_;
    float d = 0.f;
    for (int c = 0; c < C_; ++c) d += a[c] * b[c];
    atomicAdd(sep_acc, d * d);
}

// ---------------------------------------------------------------------------
// Fused main kernel: 16 query rows per workgroup (8 wave32 waves).
//   TDM ingest -> GEMM1 (dots, f32 WMMA) -> softmax/argmax/logsumexp
//   -> GEMM2 (update, f32 WMMA) -> TDM egress
// ---------------------------------------------------------------------------
__global__ void __launch_bounds__(256)
kMain(const float* __restrict__ x, const float* __restrict__ mem,
      const float* __restrict__ m2, const float* __restrict__ rm,
      float* __restrict__ fea_acc, float* __restrict__ out) {
    __shared__ float xs[ROWS * C_];      // xq tile in, update tile out: [c*16+r]
    __shared__ float ss[ROWS * SSP];     // dot scores -> attention, padded
    __shared__ float m2s[M_], rms[M_];
    __shared__ float red0[256], red1[256];
    __shared__ int   redi[256];
    __shared__ float q2s[ROWS], rqs[ROWS], maxcs[ROWS], sumcs[ROWS], maxDs[ROWS];
    __shared__ int   tops[ROWS];

    const int tid = threadIdx.x;
    const int w    = tid >> 5;           // wave id 0..7
    const int lane = tid & 31;
    const int nl   = lane & 15;          // N-column / A-row within tile
    const int hi   = lane >> 4;          // K-half selector

    const int n0 = blockIdx.x * ROWS;
    const int bb = n0 / T_;
    const int t0 = n0 % T_;
    const float* xb = x + (size_t)bb * C_ * T_ + t0;
    float*      outb = out + (size_t)bb * C_ * T_ + t0;

    // ---- Phase 1: stage xq tile (transposed 2D tile) via Tensor Data Mover
#if HAVE_TDM
    if (w == 0) {
        __builtin_amdgcn_tensor_load_to_lds(
            tdm_group0((unsigned)(uintptr_t)xs, xb), tdm_group1(),
            (i32x4){}, (i32x4){}, (i32x8){}, 0);
    }
#else
    for (int e = tid; e < ROWS * C_; e += 256) {
        int c = e >> 4, tt = e & 15;
        xs[c * 16 + tt] = xb[(size_t)c * T_ + tt];
    }
#endif
    for (int m = tid; m < M_; m += 256) { m2s[m] = m2[m]; rms[m] = rm[m]; }
#if HAVE_TDM
    if (w == 0) __builtin_amdgcn_s_wait_tensorcnt(0);
#endif
    __syncthreads();

    // ---- per-row q^2 and reciprocal norm ----------------------------------
    {
        int r = tid & 15, seg = tid >> 4;
        float s = 0.f;
        for (int j = 0; j < 32; ++j) {
            float v = xs[(seg * 32 + j) * 16 + r];
            s += v * v;
        }
        red0[tid] = s;
    }
    __syncthreads();
    if (tid < ROWS) {
        float s = 0.f;
        for (int g = 0; g < 16; ++g) s += red0[g * 16 + tid];
        q2s[tid] = s;
        rqs[tid] = 1.0f / fmaxf(sqrtf(s), EPSF);
    }
    __syncthreads();

    // ---- Phase 2: GEMM1  dots[16 x 512] = xq @ mem^T  (fp32 WMMA) ---------
    for (int mt = w; mt < M_ / 16; mt += 8) {
        const int m0 = mt * 16;
        v8f acc = {};
        for (int c0 = 0; c0 < C_; c0 += 4) {
            const int k = c0 + 2 * hi;
            v2f a, b;
            a.x = xs[k * 16 + nl];                       // A[m=nl][k]
            a.y = xs[(k + 1) * 16 + nl];                 // A[m=nl][k+1]
            const float* mr = mem + (size_t)(m0 + nl) * C_ + k;
            b.x = mr[0];                                 // B[k][n=nl]
            b.y = mr[1];                                 // B[k+1][n=nl]
            acc = __builtin_amdgcn_wmma_f32_16x16x4_f32(
                false, a, false, b, (short)0, acc, false, false);
        }
#pragma unroll
        for (int v = 0; v < 8; ++v)
            ss[(v + 8 * hi) * SSP + (m0 + nl)] = acc[v]; // dot[r][m]
    }
    __syncthreads();

    // ---- Phase 3a: row max of cos, argmax, max of D -----------------------
    const int r  = tid & 15;
    const int sg = tid >> 4;
    {
        float q2r = q2s[r], rqr = rqs[r];
        float mc = -3.4e38f, mD = -3.4e38f;
        int mi = 0;
        for (int j = 0; j < 32; ++j) {
            int m = sg * 32 + j;
            float dot = ss[r * SSP + m];
            float cs  = dot * rqr * rms[m];
            if (cs > mc) { mc = cs; mi = m; }
            float D = sqrtf(fmaxf(q2r + m2s[m] - 2.f * dot, 0.f));
            mD = fmaxf(mD, D);
        }
        red0[tid] = mc; redi[tid] = mi; red1[tid] = mD;
    }
    __syncthreads();
    if (tid < ROWS) {
        float bc = red0[tid]; int bi = redi[tid]; float bD = red1[tid];
        for (int s2 = 1; s2 < 16; ++s2) {
            float c = red0[s2 * 16 + tid]; int i2 = redi[s2 * 16 + tid];
            if (c > bc || (c == bc && i2 < bi)) { bc = c; bi = i2; }
            bD = fmaxf(bD, red1[s2 * 16 + tid]);
        }
        maxcs[tid] = bc; tops[tid] = bi; maxDs[tid] = bD;
    }
    __syncthreads();

    // ---- Phase 3b: softmax / logsumexp sums -------------------------------
    {
        float q2r = q2s[r], rqr = rqs[r];
        float mc = maxcs[r], mD = maxDs[r];
        float sc = 0.f, sD = 0.f;
        for (int j = 0; j < 32; ++j) {
            int m = sg * 32 + j;
            float dot = ss[r * SSP + m];
            sc += expf(dot * rqr * rms[m] - mc);
            float D = sqrtf(fmaxf(q2r + m2s[m] - 2.f * dot, 0.f));
            sD += expf(D - mD);
        }
        red0[tid] = sc; red1[tid] = sD;
    }
    __syncthreads();
    if (tid < ROWS) {
        float sc = 0.f, sD = 0.f;
        for (int g = 0; g < 16; ++g) { sc += red0[g * 16 + tid]; sD += red1[g * 16 + tid]; }
        sumcs[tid] = sc;
        float lse = maxDs[tid] + logf(sD);
        int top = tops[tid];
        float dt = ss[tid * SSP + top];
        float Dt = sqrtf(fmaxf(q2s[tid] + m2s[top] - 2.f * dt, 0.f));
        atomicAdd(fea_acc, (lse - Dt) * (1.0f / (float)N_));
    }
    __syncthreads();

    // ---- Phase 3c: convert scores -> attention weights in place -----------
    {
        float rqr = rqs[r], mc = maxcs[r], inv = 1.0f / sumcs[r];
        for (int j = 0; j < 32; ++j) {
            int m = sg * 32 + j;
            ss[r * SSP + m] = expf(ss[r * SSP + m] * rqr * rms[m] - mc) * inv;
        }
    }
    __syncthreads();

    // ---- Phase 4: GEMM2  update[16 x 512] = att @ mem  (fp32 WMMA) --------
    // D tiles go to the (now dead) xs buffer in TDM layout [c*16 + r].
    for (int ct = w; ct < C_ / 16; ct += 8) {
        const int c0 = ct * 16;
        v8f acc = {};
        for (int m0 = 0; m0 < M_; m0 += 4) {
            const int k = m0 + 2 * hi;
            v2f a, b;
            a.x = ss[nl * SSP + k];                      // A[r=nl][k]
            a.y = ss[nl * SSP + k + 1];
            b.x = mem[(size_t)k * C_ + c0 + nl];         // B[k][c=nl]  (coalesced)
            b.y = mem[(size_t)(k + 1) * C_ + c0 + nl];
            acc = __builtin_amdgcn_wmma_f32_16x16x4_f32(
                false, a, false, b, (short)0, acc, false, false);
        }
#pragma unroll
        for (int v = 0; v < 8; ++v) {
            int rr = v + 8 * hi;                         // query row
#if HAVE_TDM
            xs[(c0 + nl) * 16 + rr] = acc[v];
#else
            outb[(size_t)(c0 + nl) * T_ + rr] = acc[v];
#endif
        }
    }

#if HAVE_TDM
    __syncthreads();
    // ---- Phase 5: egress update tile via Tensor Data Mover ----------------
    if (w == 0) {
        __builtin_amdgcn_tensor_store_from_lds(
            tdm_group0((unsigned)(uintptr_t)xs, outb), tdm_group1(),
            (i32x4){}, (i32x4){}, (i32x8){}, 0);
        __builtin_amdgcn_s_wait_tensorcnt(0);
    }
#endif
}

// ---------------------------------------------------------------------------
// Finalize scalar losses into d_out tail.
// ---------------------------------------------------------------------------
__global__ void kFin(const float* __restrict__ fea_acc,
                     const float* __restrict__ sep_acc,
                     float* __restrict__ out) {
    out[(size_t)N_ * C_]     = *fea_acc;
    out[(size_t)N_ * C_ + 1] = sqrtf(*sep_acc) / ((float)M_ * (M_ - 1) * 2.0f);
}

extern "C" void kernel_launch(void* const* d_in, const int* in_sizes, int n_in,
                              void* d_out, int out_size, void* d_ws, size_t ws_size,
                              hipStream_t stream) {
    (void)in_sizes; (void)n_in; (void)out_size; (void)ws_size;
    const float* x   = (const float*)d_in[0];
    const float* mem = (const float*)d_in[1];
    float* out = (float*)d_out;
    float* ws  = (float*)d_ws;
    float* m2  = ws;            // 512 floats
    float* rm  = ws + 512;      // 512 floats
    float* fea = ws + 1024;     // 1 float
    float* sep = ws + 1025;     // 1 float

    kPrep<<<2, 256, 0, stream>>>(mem, m2, rm, fea, sep);
    kSep <<<(M_ * M_) / 256, 256, 0, stream>>>(mem, sep);
    kMain<<<N_ / ROWS, 256, 0, stream>>>(x, mem, m2, rm, fea, out);
    kFin <<<1, 1, 0, stream>>>(fea, sep, out);
}